// DiscriminativeLoss_1013612282343
// MI455X (gfx1250) — compile-verified
//
#include <hip/hip_runtime.h>
#include <hip/hip_bf16.h>

typedef __attribute__((ext_vector_type(2))) float v2f;
typedef __attribute__((ext_vector_type(8))) float v8f;

#define N_IMG   4
#define CHN     32
#define HW      (512 * 512)
#define NCLS    19
#define LSTRIDE 33          // 19*33 = 627 floats; col 32 holds the count
#define PPT     8           // pixels per thread
#define BLK     256

__device__ __constant__ float c_alpha = 0.05f;
__device__ __constant__ float c_beta  = 1.0f;

// ws float offsets
#define WS_SUMS   0                      // N*19*32 = 2432
#define WS_COUNTS 2432                   // N*19 = 76
#define WS_INTRAS 2512                   // 4
#define WS_INTRAV 2516                   // 4
#define WS_FLOATS 2560

static __device__ __forceinline__ v8f wmma4(v2f a, v2f b, v8f c) {
    // D = A(16x4,f32) * B(4x16,f32) + C(16x16,f32)
    return __builtin_amdgcn_wmma_f32_16x16x4_f32(
        /*neg_a=*/false, a, /*neg_b=*/false, b,
        /*c_mod=*/(short)0, c, /*reuse_a=*/false, /*reuse_b=*/false);
}

// ---------------- Pass 1: per-image class sums + counts -----------------
__global__ void __launch_bounds__(BLK)
dl_pass1(const float* __restrict__ emb, const int* __restrict__ lab,
         float* __restrict__ sums, float* __restrict__ counts) {
    __shared__ float ls[NCLS * LSTRIDE];
    const int n = blockIdx.y;
    const int t = threadIdx.x;

    for (int i = t; i < NCLS * LSTRIDE; i += BLK) ls[i] = 0.0f;
    __syncthreads();

    const float* e0 = emb + (size_t)n * CHN * HW;
    const int*   l0 = lab + (size_t)n * HW;
    const int base = blockIdx.x * (BLK * PPT) + t;

    for (int it = 0; it < PPT; ++it) {
        const int p = base + it * BLK;
        float e[CHN];
        float s = 0.0f;
#pragma unroll
        for (int c = 0; c < CHN; ++c) {
            e[c] = e0[(size_t)c * HW + p];
            s += e[c] * e[c];
        }
        const float inv = 1.0f / fmaxf(sqrtf(s), 1e-12f);
        int k = l0[p];
        if (k == 255) k = 0;
#pragma unroll
        for (int c = 0; c < CHN; ++c)
            atomicAdd(&ls[k * LSTRIDE + c], e[c] * inv);   // ds_add_f32
        atomicAdd(&ls[k * LSTRIDE + 32], 1.0f);
    }
    __syncthreads();

    for (int i = t; i < NCLS * LSTRIDE; i += BLK) {
        const int k = i / LSTRIDE, c = i % LSTRIDE;
        const float v = ls[i];
        if (c < CHN) atomicAdd(&sums[(n * NCLS + k) * CHN + c], v);
        else         atomicAdd(&counts[n * NCLS + k], v);
    }
}

// ---------------- Pass 2: intra hinge sums (embedding re-read, L2-hot) --
__global__ void __launch_bounds__(BLK)
dl_pass2(const float* __restrict__ emb, const int* __restrict__ lab,
         const float* __restrict__ sums, const float* __restrict__ counts,
         float* __restrict__ intraS, float* __restrict__ intraV) {
    __shared__ float lm[NCLS * LSTRIDE];
    __shared__ float red[16];
    const int n = blockIdx.y;
    const int t = threadIdx.x;

    for (int i = t; i < NCLS * LSTRIDE; i += BLK) {
        const int k = i / LSTRIDE, c = i % LSTRIDE;
        lm[i] = (c < CHN)
            ? sums[(n * NCLS + k) * CHN + c] / (counts[n * NCLS + k] + 1.0f)
            : 0.0f;
    }
    __syncthreads();

    const float* e0 = emb + (size_t)n * CHN * HW;
    const int*   l0 = lab + (size_t)n * HW;
    const int base = blockIdx.x * (BLK * PPT) + t;

    float S = 0.0f, V = 0.0f;
    for (int it = 0; it < PPT; ++it) {
        const int p = base + it * BLK;
        float e[CHN];
        float s = 0.0f;
#pragma unroll
        for (int c = 0; c < CHN; ++c) {
            e[c] = e0[(size_t)c * HW + p];
            s += e[c] * e[c];
        }
        const float inv = 1.0f / fmaxf(sqrtf(s), 1e-12f);
        int k = l0[p];
        if (k == 255) k = 0;
        float d = 0.0f;
#pragma unroll
        for (int c = 0; c < CHN; ++c) {
            const float diff = lm[k * LSTRIDE + c] - e[c] * inv;
            d += diff * diff;
        }
        if (k > 0 && d > c_alpha) { S += d - c_alpha; V += 1.0f; }
    }
    // wave32 reduce then cross-wave reduce
    for (int off = 16; off > 0; off >>= 1) {
        S += __shfl_down(S, off);
        V += __shfl_down(V, off);
    }
    const int w = t >> 5, lane = t & 31;
    if (lane == 0) { red[w] = S; red[8 + w] = V; }
    __syncthreads();
    if (t == 0) {
        float s = 0.0f, v = 0.0f;
        for (int i = 0; i < 8; ++i) { s += red[i]; v += red[8 + i]; }
        atomicAdd(&intraS[n], s);
        atomicAdd(&intraV[n], v);
    }
}

// ---------------- Pass 3: WMMA Gram (19x19 per image) + finalize --------
__global__ void __launch_bounds__(128)
dl_pass3(const float* __restrict__ sums, const float* __restrict__ counts,
         const float* __restrict__ intraS, const float* __restrict__ intraV,
         float* __restrict__ out) {
    __shared__ float lm[N_IMG][32][LSTRIDE];   // padded means (rows>=19 zero)
    __shared__ float lg[N_IMG][32][LSTRIDE];   // Gram results
    __shared__ float lint[N_IMG];
    const int t = threadIdx.x;
    const int w = t >> 5, lane = t & 31;       // wave w handles image w

    // fill padded means: 4*32*33 = 4224 = 128 * 33 (uniform, no divergence)
    for (int i = t; i < N_IMG * 32 * LSTRIDE; i += 128) {
        const int n = i / (32 * LSTRIDE);
        const int r = (i / LSTRIDE) % 32;
        const int c = i % LSTRIDE;
        float v = 0.0f;
        if (c < CHN && r < NCLS)
            v = sums[(n * NCLS + r) * CHN + c] / (counts[n * NCLS + r] + 1.0f);
        lm[n][r][c] = v;
    }
    __syncthreads();

    // G = M * M^T  (M padded 32x32), tiles 2x2 of 16x16, K=32 via 8x K4.
    // A 16x4 f32 layout: lanes 0-15 rows, K split across lane halves.
    // B 4x16 gather mirrors it; for the symmetric Gram both gathers coincide.
    v8f acc00 = {}, acc01 = {}, acc10 = {}, acc11 = {};
    const int row = lane & 15;
    const int kb  = (lane >> 4) << 1;
#pragma unroll
    for (int k0 = 0; k0 < 32; k0 += 4) {
        v2f a0, a1;
        a0.x = lm[w][row][k0 + kb];
        a0.y = lm[w][row][k0 + kb + 1];
        a1.x = lm[w][16 + row][k0 + kb];
        a1.y = lm[w][16 + row][k0 + kb + 1];
        acc00 = wmma4(a0, a0, acc00);
        acc01 = wmma4(a0, a1, acc01);
        acc10 = wmma4(a1, a0, acc10);
        acc11 = wmma4(a1, a1, acc11);
    }
    // C/D layout: VGPR r -> M = r (lanes 0-15) or r+8 (lanes 16-31), N = lane&15
#pragma unroll
    for (int r = 0; r < 8; ++r) {
        const int M = (lane < 16) ? r : (r + 8);
        const int Nc = lane & 15;
        lg[w][M][Nc]           = acc00[r];
        lg[w][M][16 + Nc]      = acc01[r];
        lg[w][16 + M][Nc]      = acc10[r];
        lg[w][16 + M][16 + Nc] = acc11[r];
    }
    __syncthreads();

    // per-image inter term: D_ij = G_ii + G_jj - 2 G_ij, presence-masked
    float num = 0.0f, den = 0.0f;
    for (int p = lane; p < NCLS * NCLS; p += 32) {
        const int i = p / NCLS;
        const int j = p - i * NCLS;
        const float pi = (counts[w * NCLS + i] > 0.0f) ? 1.0f : 0.0f;
        const float pj = (counts[w * NCLS + j] > 0.0f) ? 1.0f : 0.0f;
        const float D = (lg[w][i][i] + lg[w][j][j] - 2.0f * lg[w][i][j]) * pi * pj;
        const bool im = (D < c_beta) && (D > 0.0f);
        if (i >= 1 && j >= 1 && im) num += c_beta - D;
        if (i >= 1 && im)           den += 1.0f;
    }
    for (int off = 16; off > 0; off >>= 1) {
        num += __shfl_down(num, off);
        den += __shfl_down(den, off);
    }
    if (lane == 0) lint[w] = num / (den + 1.0f) * 0.5f;
    __syncthreads();

    if (t == 0) {
        float inter = 0.0f, intra = 0.0f;
        for (int n = 0; n < N_IMG; ++n) {
            inter += lint[n];
            intra += intraS[n] / (intraV[n] + 1.0f);
        }
        out[0] = intra * 0.25f;   // mean over N
        out[1] = inter * 0.25f;
    }
}

extern "C" void kernel_launch(void* const* d_in, const int* in_sizes, int n_in,
                              void* d_out, int out_size, void* d_ws, size_t ws_size,
                              hipStream_t stream) {
    const float* emb = (const float*)d_in[0];
    const int*   lab = (const int*)d_in[1];
    float* ws     = (float*)d_ws;
    float* sums   = ws + WS_SUMS;
    float* counts = ws + WS_COUNTS;
    float* intraS = ws + WS_INTRAS;
    float* intraV = ws + WS_INTRAV;

    hipMemsetAsync(d_ws, 0, WS_FLOATS * sizeof(float), stream);

    dim3 grid(HW / (BLK * PPT), N_IMG);   // 128 x 4
    dl_pass1<<<grid, BLK, 0, stream>>>(emb, lab, sums, counts);
    dl_pass2<<<grid, BLK, 0, stream>>>(emb, lab, sums, counts, intraS, intraV);
    dl_pass3<<<1, 128, 0, stream>>>(sums, counts, intraS, intraV, (float*)d_out);
}